// CEOLoss_71803263255030
// MI455X (gfx1250) — compile-verified
//
#include <hip/hip_runtime.h>

#define BTOT  4194304
#define NBLK  1024
#define NTHR  256
#define WAVES (NTHR / 32)
// (BTOT/4) float4 elements / (NBLK*NTHR) threads = 4 iterations per thread
#define ITERS ((BTOT / 4) / (NBLK * NTHR))

#define LOG2E 1.4426950408889634f
#define LN2   0.69314718055994531f

typedef __attribute__((ext_vector_type(2))) float v2f;
typedef __attribute__((ext_vector_type(8))) float v8f;

// nll(x, c) = |x - (c-3)| + log( sum_{l=-3..3} exp(-|x - l|) )
// With r = clamp(rint(x),-3,3), f = x - r, the 7 terms split into two
// geometric tails of ratio e^-1:
//   S = exp(-|f|) + exp(f)*G[3-r] + exp(-f)*G[3+r],  G[n] = sum_{k=1..n} e^-k
// S is always in (~0.08, 2.11): safe for raw v_exp_f32 / v_log_f32.
__device__ __forceinline__ float nll_elem(float x, int c, const float* gtab) {
    float r  = __builtin_amdgcn_fmed3f(rintf(x), -3.0f, 3.0f); // v_rndne + v_med3
    float t  = (x - r) * LOG2E;
    float ea = __builtin_amdgcn_exp2f(t);                      // v_exp_f32
    float eb = __builtin_amdgcn_exp2f(-t);                     // v_exp_f32 (neg = src mod)
    int ilo  = (int)r + 3;                                     // n_lo in [0,6]
    float Glo = gtab[ilo];                                     // ds_load_b32
    float Ghi = gtab[6 - ilo];                                 // ds_load_b32
    float S   = fminf(ea, eb) + ea * Ghi + eb * Glo;           // min + 2x v_fma
    float w   = x - (float)c + 3.0f;                           // x - (c-3)
    // log(S) + |w|  ==  fma(ln2, log2(S), |w|)
    return fmaf(LN2, __builtin_amdgcn_logf(S), fabsf(w));
}

__global__ void __launch_bounds__(NTHR)
ceol_partial(const float4* __restrict__ x4, const int4* __restrict__ c4,
             float* __restrict__ partials) {
    __shared__ float gtab[8];
    __shared__ float wsum[2 * WAVES];

    if (threadIdx.x == 0) {
        gtab[0] = 0.0f;
        gtab[1] = 0.36787944117144233f;   // e^-1
        gtab[2] = 0.50321472440805498f;   // + e^-2
        gtab[3] = 0.55300179277591887f;   // + e^-3
        gtab[4] = 0.57131743166465310f;   // + e^-4
        gtab[5] = 0.57805537866373862f;   // + e^-5
        gtab[6] = 0.58053413084040506f;   // + e^-6
        gtab[7] = 0.0f;
    }
    __syncthreads();

    const unsigned tid    = blockIdx.x * NTHR + threadIdx.x;
    const unsigned stride = NBLK * NTHR;

    float acc = 0.0f;
#pragma unroll
    for (int it = 0; it < ITERS; ++it) {
        unsigned idx = tid + (unsigned)it * stride;   // coalesced: 16B/lane
        float4 xv = x4[idx];
        int4   cv = c4[idx];
        acc += nll_elem(xv.x, cv.x, gtab);
        acc += nll_elem(xv.y, cv.y, gtab);
        acc += nll_elem(xv.z, cv.z, gtab);
        acc += nll_elem(xv.w, cv.w, gtab);
    }

    // ---- wave32 reduction via the matrix unit ----
    // A (16x4): lane L<16 -> (M=L, K=0)=acc; lane L>=16 -> (M=L-16, K=2)=acc.
    // B (4x16) = ones  =>  D[m,n] = acc[m] + acc[m+16]  (replicated over n).
    // EXEC is all-ones here (no divergence since the barrier).
    v2f a; a.x = acc;  a.y = 0.0f;
    v2f b; b.x = 1.0f; b.y = 1.0f;
    v8f cz = {};
    v8f dm = __builtin_amdgcn_wmma_f32_16x16x4_f32(false, a, false, b, (short)0,
                                                   cz, false, false);
    // lanes 0-15: dm holds rows 0-7 ; lanes 16-31: rows 8-15.
    float s = dm[0] + dm[1] + dm[2] + dm[3] + dm[4] + dm[5] + dm[6] + dm[7];

    const unsigned lane = threadIdx.x & 31u;
    const unsigned wid  = threadIdx.x >> 5;
    // lane 0 stores rows0-7 half, lane 16 stores rows8-15 half: no shuffle needed
    if ((lane & 15u) == 0u) wsum[(wid << 1) | (lane >> 4)] = s;
    __syncthreads();

    if (threadIdx.x == 0) {
        float bs = 0.0f;
#pragma unroll
        for (int w = 0; w < 2 * WAVES; ++w) bs += wsum[w];
        partials[blockIdx.x] = bs;       // fixed order -> deterministic
    }
}

__global__ void __launch_bounds__(NTHR)
ceol_final(const float* __restrict__ partials, float* __restrict__ out) {
    __shared__ float sm[NTHR];
    float v = 0.0f;
    for (int i = threadIdx.x; i < NBLK; i += NTHR) v += partials[i];
    sm[threadIdx.x] = v;
    __syncthreads();
#pragma unroll
    for (int off = NTHR / 2; off > 0; off >>= 1) {
        if (threadIdx.x < off) sm[threadIdx.x] += sm[threadIdx.x + off];
        __syncthreads();
    }
    if (threadIdx.x == 0) out[0] = sm[0] * (1.0f / (float)BTOT);
}

extern "C" void kernel_launch(void* const* d_in, const int* in_sizes, int n_in,
                              void* d_out, int out_size, void* d_ws, size_t ws_size,
                              hipStream_t stream) {
    // inputs: 0=x (f32,B), 1=y (dead), 2=logits_4cls (dead), 3=class_y (i32,B)
    const float4* x4 = (const float4*)d_in[0];
    const int4*   c4 = (const int4*)d_in[3];
    float* partials  = (float*)d_ws;             // NBLK floats = 4 KB scratch

    ceol_partial<<<NBLK, NTHR, 0, stream>>>(x4, c4, partials);
    ceol_final<<<1, NTHR, 0, stream>>>(partials, (float*)d_out);
}